// GNNLayer_12524124635909
// MI455X (gfx1250) — compile-verified
//
#include <hip/hip_runtime.h>
#include <hip/hip_bf16.h>

// ---------------------------------------------------------------------------
// GAT layer (single head) + BatchNorm for MI455X (gfx1250, wave32, WMMA).
//   N = 50000 nodes, E = 800000 edges (+N self loops), C = 128.
// Stage 1 GEMM uses V_WMMA_F32_16X16X4_F32 (exact fp32 matrix math).
// ---------------------------------------------------------------------------

#define C_DIM 128
#define WAVE  32

typedef float v2f __attribute__((ext_vector_type(2)));
typedef float v8f __attribute__((ext_vector_type(8)));

// --- float atomic add via hardware global_atomic_add_f32 --------------------
__device__ __forceinline__ void atomAddF(float* p, float v) {
    unsafeAtomicAdd(p, v);
}

// --- monotone float<->uint encoding for atomic float max --------------------
__device__ __forceinline__ unsigned encF(float f) {
    unsigned u = __float_as_uint(f);
    return (u & 0x80000000u) ? ~u : (u | 0x80000000u);
}
__device__ __forceinline__ float decF(unsigned u) {
    return (u & 0x80000000u) ? __uint_as_float(u & 0x7fffffffu)
                             : __uint_as_float(~u);
}

// ---------------------------------------------------------------------------
// K0: zero fill
// ---------------------------------------------------------------------------
__global__ void k_zero(float* __restrict__ p, long n) {
    long i = (long)blockIdx.x * blockDim.x + threadIdx.x;
    long stride = (long)gridDim.x * blockDim.x;
    for (; i < n; i += stride) p[i] = 0.0f;
}

// ---------------------------------------------------------------------------
// K1: xl = x @ W^T   (N x 128) = (N x 128)(128 x 128)^T  via fp32 WMMA.
// One wave per 16-row band; each wave computes the full 128 output channels
// (8 tiles of 16) with v_wmma_f32_16x16x4_f32, K stepped by 4.
// W is staged transposed in LDS: wt[k][o] = W[o][k] (64 KB).
// ---------------------------------------------------------------------------
__global__ __launch_bounds__(256) void k_gemm_wmma(
        const float* __restrict__ X, const float* __restrict__ W,
        float* __restrict__ XL, int N) {
    __shared__ float wt[C_DIM * C_DIM];   // 64 KB, wt[k*128 + o] = W[o*128 + k]
    for (int i = threadIdx.x; i < C_DIM * C_DIM; i += 256) {
        int o = i & (C_DIM - 1);
        int k = i >> 7;
        wt[k * C_DIM + o] = W[o * C_DIM + k];   // conflict-free LDS writes
    }
    __syncthreads();

    const int lane = threadIdx.x & (WAVE - 1);
    const int wid  = threadIdx.x >> 5;
    const int band = blockIdx.x * 8 + wid;          // 16-row band index
    if (band * 16 >= N) return;                     // wave-uniform exit

    const int m0    = band * 16;
    const int mrow  = lane & 15;                    // A: M = lane (0..15)
    const int khalf = (lane >> 4) * 2;              // A/B: K pair select (0 or 2)
    const float* xrow = X + (long)(m0 + mrow) * C_DIM;

    v8f acc[8] = {};                                // 8 column tiles of 16

    for (int k0 = 0; k0 < C_DIM; k0 += 4) {
        // A tile 16x4: lane holds (K=khalf, K=khalf+1) of row mrow
        v2f a = *(const v2f*)(xrow + k0 + khalf);
#pragma unroll
        for (int t = 0; t < 8; ++t) {
            // B tile 4x16: B[k][n] = W[t*16+n][k0+k] = wt[(k0+k)*128 + t*16+n]
            const int ocol = t * 16 + mrow;
            v2f b;
            b.x = wt[(k0 + khalf)     * C_DIM + ocol];
            b.y = wt[(k0 + khalf + 1) * C_DIM + ocol];
            acc[t] = __builtin_amdgcn_wmma_f32_16x16x4_f32(
                false, a, false, b, (short)0, acc[t], false, false);
        }
    }

    // C/D layout: VGPR r -> lanes 0-15: (M=r, N=lane); lanes 16-31: (M=r+8)
    const int ncol  = lane & 15;
    const int rbase = (lane >> 4) * 8;
#pragma unroll
    for (int t = 0; t < 8; ++t)
#pragma unroll
        for (int r = 0; r < 8; ++r)
            XL[(long)(m0 + rbase + r) * C_DIM + t * 16 + ncol] = acc[t][r];
}

// ---------------------------------------------------------------------------
// K2: per-node attention scalars (one wave per node, 4 channels per lane)
//   a_i[n] = <xl[n],att_i> + <emb[n],att_em_i>
//   a_j[n] = <xl[n],att_j> + <emb[n],att_em_j>
// ---------------------------------------------------------------------------
__global__ __launch_bounds__(256) void k_node_alpha(
        const float* __restrict__ xl, const float* __restrict__ emb,
        const float* __restrict__ att_i, const float* __restrict__ att_j,
        const float* __restrict__ att_em_i, const float* __restrict__ att_em_j,
        float* __restrict__ a_i, float* __restrict__ a_j, int N) {
    const int lane = threadIdx.x & (WAVE - 1);
    const int node = blockIdx.x * 8 + (threadIdx.x >> 5);
    if (node >= N) return;                          // wave-uniform

    const float4 xv  = *(const float4*)(xl  + (long)node * C_DIM + lane * 4);
    const float4 ev  = *(const float4*)(emb + (long)node * C_DIM + lane * 4);
    const float4 ai  = *(const float4*)(att_i    + lane * 4);
    const float4 aj  = *(const float4*)(att_j    + lane * 4);
    const float4 aei = *(const float4*)(att_em_i + lane * 4);
    const float4 aej = *(const float4*)(att_em_j + lane * 4);

    float si = xv.x*ai.x + xv.y*ai.y + xv.z*ai.z + xv.w*ai.w
             + ev.x*aei.x + ev.y*aei.y + ev.z*aei.z + ev.w*aei.w;
    float sj = xv.x*aj.x + xv.y*aj.y + xv.z*aj.z + xv.w*aj.w
             + ev.x*aej.x + ev.y*aej.y + ev.z*aej.z + ev.w*aej.w;
#pragma unroll
    for (int off = 16; off > 0; off >>= 1) {
        si += __shfl_xor(si, off, WAVE);
        sj += __shfl_xor(sj, off, WAVE);
    }
    if (lane == 0) { a_i[node] = si; a_j[node] = sj; }
}

// ---------------------------------------------------------------------------
// K3: per-edge logits + leaky ReLU + segment max (atomic on encoded uint)
// Edges e < E come from edge_index; e >= E are self loops (src=dst=e-E).
// ---------------------------------------------------------------------------
__global__ void k_edge_max(const int* __restrict__ ei,
                           const float* __restrict__ a_i,
                           const float* __restrict__ a_j,
                           float* __restrict__ alpha,
                           unsigned* __restrict__ amax, int E, int EN) {
    int e = blockIdx.x * 256 + threadIdx.x;
    if (e >= EN) return;
    int s = (e < E) ? ei[e]     : (e - E);
    int d = (e < E) ? ei[E + e] : (e - E);
    float al = a_i[d] + a_j[s];
    al = (al >= 0.0f) ? al : 0.2f * al;             // leaky_relu(0.2)
    alpha[e] = al;
    atomicMax(&amax[d], encF(al));
}

// ---------------------------------------------------------------------------
// K4: exp(alpha - max[dst]) and segment sum (denominator)
// ---------------------------------------------------------------------------
__global__ void k_edge_exp(const int* __restrict__ ei,
                           const unsigned* __restrict__ amax,
                           float* __restrict__ alpha,   // in: logit, out: exp
                           float* __restrict__ denom, int E, int EN) {
    int e = blockIdx.x * 256 + threadIdx.x;
    if (e >= EN) return;
    int d = (e < E) ? ei[E + e] : (e - E);
    float ex = expf(alpha[e] - decF(amax[d]));
    alpha[e] = ex;
    atomAddF(&denom[d], ex);
}

// ---------------------------------------------------------------------------
// K5: aggregation  out[dst] += (ex/denom[dst]) * xl[src]
// One wave per edge, 4 channels per lane (float4 gather + 4 f32 atomics).
// ---------------------------------------------------------------------------
__global__ __launch_bounds__(256) void k_aggregate(
        const int* __restrict__ ei, const float* __restrict__ xl,
        const float* __restrict__ exw, const float* __restrict__ denom,
        float* __restrict__ out, int E, int EN) {
    const int lane = threadIdx.x & (WAVE - 1);
    const int e    = blockIdx.x * 8 + (threadIdx.x >> 5);
    if (e >= EN) return;                            // wave-uniform
    int s = (e < E) ? ei[e]     : (e - E);
    int d = (e < E) ? ei[E + e] : (e - E);
    float w = exw[e] / (denom[d] + 1e-16f);
    const float4 xj = *(const float4*)(xl + (long)s * C_DIM + lane * 4);
    float* op = out + (long)d * C_DIM + lane * 4;
    atomAddF(op + 0, w * xj.x);
    atomAddF(op + 1, w * xj.y);
    atomAddF(op + 2, w * xj.z);
    atomAddF(op + 3, w * xj.w);
}

// ---------------------------------------------------------------------------
// K6: add bias in place + per-channel sum / sumsq partials (BN stats)
// blockDim = 128 (thread == channel); block-local partials, 1 atomic each.
// ---------------------------------------------------------------------------
__global__ __launch_bounds__(128) void k_bias_stats(
        float* __restrict__ out, const float* __restrict__ bias,
        float* __restrict__ stats, int N) {
    const int c = threadIdx.x;
    const float b = bias[c];
    float s = 0.0f, q = 0.0f;
    for (int r = blockIdx.x; r < N; r += gridDim.x) {
        float v = out[(long)r * C_DIM + c] + b;
        out[(long)r * C_DIM + c] = v;
        s += v; q += v * v;
    }
    atomAddF(&stats[c],           s);
    atomAddF(&stats[C_DIM + c],   q);
}

// ---------------------------------------------------------------------------
// K7: fold stats -> per-channel scale/shift  (biased batch variance)
// ---------------------------------------------------------------------------
__global__ __launch_bounds__(128) void k_bn_params(
        const float* __restrict__ stats, const float* __restrict__ gamma,
        const float* __restrict__ beta, float* __restrict__ bnp, int N) {
    const int c = threadIdx.x;
    float inv_n = 1.0f / (float)N;
    float mu  = stats[c] * inv_n;
    float var = stats[C_DIM + c] * inv_n - mu * mu;
    float sc  = gamma[c] * rsqrtf(var + 1e-5f);
    bnp[c]         = sc;
    bnp[C_DIM + c] = beta[c] - mu * sc;
}

// ---------------------------------------------------------------------------
// K8: apply BN in place
// ---------------------------------------------------------------------------
__global__ void k_bn_apply(float* __restrict__ out,
                           const float* __restrict__ bnp, long total) {
    long i = (long)blockIdx.x * blockDim.x + threadIdx.x;
    long stride = (long)gridDim.x * blockDim.x;
    for (; i < total; i += stride) {
        int c = (int)(i & (C_DIM - 1));
        out[i] = out[i] * bnp[c] + bnp[C_DIM + c];
    }
}

// ---------------------------------------------------------------------------
extern "C" void kernel_launch(void* const* d_in, const int* in_sizes, int n_in,
                              void* d_out, int out_size, void* d_ws, size_t ws_size,
                              hipStream_t stream) {
    const float* x        = (const float*)d_in[0];
    const int*   ei       = (const int*)  d_in[1];   // [2,E] int32: src row 0, dst row 1
    const float* emb      = (const float*)d_in[2];
    const float* lin_w    = (const float*)d_in[3];
    const float* att_i    = (const float*)d_in[4];
    const float* att_j    = (const float*)d_in[5];
    const float* att_em_i = (const float*)d_in[6];
    const float* att_em_j = (const float*)d_in[7];
    const float* bias     = (const float*)d_in[8];
    const float* gamma    = (const float*)d_in[9];
    const float* beta     = (const float*)d_in[10];
    float* out = (float*)d_out;

    const int N  = in_sizes[0] / C_DIM;
    const int E  = in_sizes[1] / 2;
    const int EN = E + N;                           // edges + self loops

    // workspace layout (floats)
    float*    ws    = (float*)d_ws;
    float*    xl    = ws;                           // N*128
    float*    a_i   = xl + (long)N * C_DIM;         // N
    float*    a_j   = a_i + N;                      // N
    unsigned* amax  = (unsigned*)(a_j + N);         // N  (encoded floats)
    float*    denom = (float*)(amax + N);           // N
    float*    exarr = denom + N;                    // EN (logit -> exp, in place)
    float*    stats = exarr + EN;                   // 256 (sum, sumsq)
    float*    bnp   = stats + 256;                  // 256 (scale, shift)

    // K0: deterministic zero init of accumulators
    k_zero<<<4096, 256, 0, stream>>>(out, (long)N * C_DIM);
    k_zero<<<256,  256, 0, stream>>>((float*)amax, 2L * N);   // amax enc + denom
    k_zero<<<1,    256, 0, stream>>>(stats, 256);

    // K1: GEMM via fp32 WMMA (N is an exact multiple of 16: 50000 = 3125*16)
    const int bands = (N + 15) / 16;
    k_gemm_wmma<<<(bands + 7) / 8, 256, 0, stream>>>(x, lin_w, xl, N);

    // K2: per-node attention scalars
    k_node_alpha<<<(N + 7) / 8, 256, 0, stream>>>(xl, emb, att_i, att_j,
                                                  att_em_i, att_em_j, a_i, a_j, N);

    // K3/K4: segment softmax over destinations
    k_edge_max<<<(EN + 255) / 256, 256, 0, stream>>>(ei, a_i, a_j, exarr, amax, E, EN);
    k_edge_exp<<<(EN + 255) / 256, 256, 0, stream>>>(ei, amax, exarr, denom, E, EN);

    // K5: weighted scatter-add aggregation
    k_aggregate<<<(EN + 7) / 8, 256, 0, stream>>>(ei, xl, exarr, denom, out, E, EN);

    // K6-K8: bias + batchnorm (training-mode, biased variance)
    k_bias_stats<<<256, 128, 0, stream>>>(out, bias, stats, N);
    k_bn_params<<<1, 128, 0, stream>>>(stats, gamma, beta, bnp, N);
    k_bn_apply<<<4096, 256, 0, stream>>>(out, bnp, (long)N * C_DIM);
}